// SequenceProgramDecoder_70265664962663
// MI455X (gfx1250) — compile-verified
//
#include <hip/hip_runtime.h>
#include <math.h>

// Problem constants from the reference
#define B_  64
#define S_  256
#define H_  1024
#define V_  10000
#define H2_ 2048
#define H3_ 3072

typedef __attribute__((ext_vector_type(2))) float v2f;
typedef __attribute__((ext_vector_type(8))) float v8f;

__device__ __forceinline__ float sigmoidf_(float x) { return 1.0f / (1.0f + expf(-x)); }

// -------------------------------------------------------------------------
// Generic fp32 WMMA GEMM:  C[M,N] = act( A[M,K] @ Bt[N,K]^T + bias[N] )
// One wave (32 threads) per 16x16 output tile, K consumed 4 at a time via
// V_WMMA_F32_16X16X4_F32 (exact fp32 matrix-core math, matches reference).
// Optional row gather on A (used to fuse the embedding lookup into the GEMM).
//
// Fragment layouts per CDNA5 ISA 7.12.2:
//   A (16x4):  lanes 0-15 = rows M0..M15 holding K{0,1}; lanes 16-31 same rows, K{2,3}
//   B (4x16):  lanes 0-15 = cols N0..N15 holding K{0,1}; lanes 16-31 same cols, K{2,3}
//   C (16x16): VGPR g: lanes 0-15 -> M=m0+g, lanes 16-31 -> M=m0+8+g; N = n0+(lane&15)
// -------------------------------------------------------------------------
__global__ void gemm_wmma_f32(const float* __restrict__ A, int lda,
                              const float* __restrict__ Bt, int ldb,
                              const float* __restrict__ bias,
                              float* __restrict__ C, int ldc,
                              int K, int act_tanh,
                              const int* __restrict__ gather)
{
    const int lane  = threadIdx.x & 31;
    const int n0    = blockIdx.x * 16;
    const int m0    = blockIdx.y * 16;
    const int lhalf = lane >> 4;       // 0 or 1
    const int l15   = lane & 15;
    const int kbase = lhalf << 1;      // 0 or 2

    int arow_idx = m0 + l15;
    if (gather) arow_idx = gather[arow_idx];
    const float* arow = A  + (long)arow_idx   * lda;
    const float* brow = Bt + (long)(n0 + l15) * ldb;

    v8f acc = {};
    #pragma unroll 4
    for (int k = 0; k < K; k += 4) {
        v2f a, b;
        a.x = arow[k + kbase];
        a.y = arow[k + kbase + 1];
        b.x = brow[k + kbase];
        b.y = brow[k + kbase + 1];
        // 8-arg form: (neg_a, A, neg_b, B, c_mod, C, reuse_a, reuse_b)
        acc = __builtin_amdgcn_wmma_f32_16x16x4_f32(false, a, false, b,
                                                    (short)0, acc, false, false);
    }

    const int   col    = n0 + l15;
    const int   rowoff = m0 + (lhalf << 3);
    const float bv     = bias ? bias[col] : 0.0f;
    #pragma unroll
    for (int g = 0; g < 8; ++g) {
        float v = acc[g] + bv;
        if (act_tanh) v = tanhf(v);
        C[(long)(rowoff + g) * ldc + col] = v;
    }
}

// -------------------------------------------------------------------------
// Scores-fusion precompute: u[k] = sum_h v_attn[h] * W_attn[h,k]  (k < 2H),
// c0 = v_attn . b_attn.  Collapses the (B,S,2H)x(H,2H) "energy" GEMM.
// -------------------------------------------------------------------------
__global__ void attn_u_kernel(const float* __restrict__ W_attn,
                              const float* __restrict__ b_attn,
                              const float* __restrict__ v_attn,
                              float* __restrict__ u, float* __restrict__ c0)
{
    if (blockIdx.x < 8) {
        int k = blockIdx.x * 256 + threadIdx.x;   // 0..2047, coalesced over k
        float acc = 0.0f;
        for (int h = 0; h < H_; ++h) acc += v_attn[h] * W_attn[(long)h * H2_ + k];
        u[k] = acc;
    } else if (threadIdx.x == 0) {
        float acc = 0.0f;
        for (int h = 0; h < H_; ++h) acc += v_attn[h] * b_attn[h];
        c0[0] = acc;
    }
}

// -------------------------------------------------------------------------
// Grid-wide barrier (persistent-kernel style), agent-scope atomics on ws.
// Counter is zeroed by reset_sync_kernel before each launch -> graph safe.
// -------------------------------------------------------------------------
__global__ void reset_sync_kernel(unsigned* sync) { if (threadIdx.x == 0) sync[0] = 0u; }

__device__ __forceinline__ void grid_sync(unsigned* cnt, unsigned target)
{
    __syncthreads();
    if (threadIdx.x == 0) {
        __hip_atomic_fetch_add(cnt, 1u, __ATOMIC_ACQ_REL, __HIP_MEMORY_SCOPE_AGENT);
        while (__hip_atomic_load(cnt, __ATOMIC_ACQUIRE, __HIP_MEMORY_SCOPE_AGENT) < target)
            __builtin_amdgcn_s_sleep(1);
    }
    __syncthreads();
}

// -------------------------------------------------------------------------
// GRU scan: 64 sequential steps. gi (x @ W_ih^T + b_ih) is precomputed by the
// WMMA GEMM; only gh = h @ W_hh^T is sequential. Each h_new[i] is owned by an
// 8-lane subgroup that computes all three W_hh row-dots (r,z,n), reduces via
// ds-bpermute shuffles, applies gate math, ping-pongs h, one grid sync/step.
// 32 blocks x 256 thr = 8192 = 1024 outputs x 8 lanes.
// -------------------------------------------------------------------------
__global__ void gru_scan_kernel(const float* __restrict__ gi,
                                const float* __restrict__ W_hh,
                                const float* __restrict__ b_hh,
                                const float* __restrict__ h0,
                                float* __restrict__ hbuf,      // 2 x H
                                float* __restrict__ rnn,       // B x H
                                float* __restrict__ hidden_out,
                                unsigned* __restrict__ sync)
{
    const int t  = blockIdx.x * blockDim.x + threadIdx.x;  // 0..8191
    const int i  = t >> 3;                                 // hidden index
    const int l  = t & 7;                                  // lane-in-group
    const unsigned nb = gridDim.x;
    unsigned sync_no = 0;

    if (t < H_) hbuf[t] = h0[t];
    grid_sync(sync, (++sync_no) * nb);

    const float* wr = W_hh + (long)i          * H_;
    const float* wz = W_hh + (long)(H_  + i)  * H_;
    const float* wn = W_hh + (long)(2*H_ + i) * H_;

    for (int step = 0; step < B_; ++step) {
        const float* cur = hbuf + (step & 1) * H_;
        float*       nxt = hbuf + ((step + 1) & 1) * H_;

        float sr = 0.0f, sz = 0.0f, sn = 0.0f;
        for (int k = l; k < H_; k += 8) {
            float hv = cur[k];
            sr += hv * wr[k];
            sz += hv * wz[k];
            sn += hv * wn[k];
        }
        #pragma unroll
        for (int off = 4; off >= 1; off >>= 1) {
            sr += __shfl_down(sr, off, 8);
            sz += __shfl_down(sz, off, 8);
            sn += __shfl_down(sn, off, 8);
        }
        if (l == 0) {
            const float* g = gi + (long)step * H3_;
            float r = sigmoidf_(g[i]        + b_hh[i]        + sr);
            float z = sigmoidf_(g[H_ + i]   + b_hh[H_ + i]   + sz);
            float n = tanhf    (g[2*H_ + i] + r * (b_hh[2*H_ + i] + sn));
            float hnew = (1.0f - z) * n + z * cur[i];
            nxt[i] = hnew;
            rnn[(long)step * H_ + i] = hnew;
        }
        __threadfence();
        grid_sync(sync, (++sync_no) * nb);
    }
    // B_=64 even -> final h is hbuf[0..H)
    if (t < H_) hidden_out[t] = hbuf[t];
}

// t1[b] = u[0:H] . rnn_output[b]
__global__ void t1_kernel(const float* __restrict__ u, const float* __restrict__ rnn,
                          float* __restrict__ t1)
{
    int b = blockIdx.x, lane = threadIdx.x;
    float acc = 0.0f;
    for (int h = lane; h < H_; h += 32) acc += u[h] * rnn[(long)b * H_ + h];
    #pragma unroll
    for (int off = 16; off >= 1; off >>= 1) acc += __shfl_down(acc, off, 32);
    if (lane == 0) t1[b] = acc;
}

// scores[b,s] = t1[b] + u2 . enc[s,b,:] + c0   (one wave per (b,s), coalesced)
__global__ void scores_kernel(const float* __restrict__ u2, const float* __restrict__ enc,
                              const float* __restrict__ t1, const float* __restrict__ c0,
                              float* __restrict__ scores)
{
    int p    = blockIdx.x * 8 + (threadIdx.x >> 5);  // 0..B*S-1
    int lane = threadIdx.x & 31;
    int b = p >> 8, s = p & 255;
    const float* e = enc + ((long)s * B_ + b) * H_;
    float acc = 0.0f;
    for (int h = lane; h < H_; h += 32) acc += u2[h] * e[h];
    #pragma unroll
    for (int off = 16; off >= 1; off >>= 1) acc += __shfl_down(acc, off, 32);
    if (lane == 0) scores[b * S_ + s] = acc + t1[b] + c0[0];
}

// Row softmax over S=256 (== blockDim), LDS tree reductions.
__global__ void softmax_kernel(const float* __restrict__ scores,
                               float* __restrict__ attnw, float* __restrict__ attn_out)
{
    __shared__ float sm[256];
    int b = blockIdx.x, s = threadIdx.x;
    float v = scores[b * S_ + s];
    sm[s] = v; __syncthreads();
    for (int off = 128; off >= 1; off >>= 1) {
        if (s < off) sm[s] = fmaxf(sm[s], sm[s + off]);
        __syncthreads();
    }
    float m = sm[0]; __syncthreads();
    float e = expf(v - m);
    sm[s] = e; __syncthreads();
    for (int off = 128; off >= 1; off >>= 1) {
        if (s < off) sm[s] += sm[s + off];
        __syncthreads();
    }
    float w = e / sm[0];
    attnw[b * S_ + s]    = w;
    attn_out[b * S_ + s] = w;
}

// context[b,h] = sum_s w[b,s] * enc[s,b,h]; also assembles concat_input rows.
__global__ void context_kernel(const float* __restrict__ attnw, const float* __restrict__ enc,
                               const float* __restrict__ rnn, float* __restrict__ cin)
{
    int b = blockIdx.x >> 2;
    int h = (blockIdx.x & 3) * 256 + threadIdx.x;
    float acc = 0.0f;
    for (int s = 0; s < S_; ++s)
        acc += attnw[b * S_ + s] * enc[((long)s * B_ + b) * H_ + h];
    cin[(long)b * H2_ + H_ + h] = acc;                  // context half
    cin[(long)b * H2_ + h]      = rnn[(long)b * H_ + h]; // rnn half
}

// -------------------------------------------------------------------------
extern "C" void kernel_launch(void* const* d_in, const int* in_sizes, int n_in,
                              void* d_out, int out_size, void* d_ws, size_t ws_size,
                              hipStream_t stream)
{
    const int*   input_seq   = (const int*)d_in[0];
    const float* last_hidden = (const float*)d_in[1];
    const float* enc         = (const float*)d_in[2];
    const float* emb         = (const float*)d_in[3];
    const float* W_ih        = (const float*)d_in[4];
    const float* W_hh        = (const float*)d_in[5];
    const float* b_ih        = (const float*)d_in[6];
    const float* b_hh        = (const float*)d_in[7];
    const float* W_attn      = (const float*)d_in[8];
    const float* b_attn      = (const float*)d_in[9];
    const float* v_attn      = (const float*)d_in[10];
    const float* W_concat    = (const float*)d_in[11];
    const float* b_concat    = (const float*)d_in[12];
    const float* W_out       = (const float*)d_in[13];
    const float* b_out       = (const float*)d_in[14];

    float* ws   = (float*)d_ws;
    float* u    = ws;                 // 2048
    float* c0   = ws + 2048;          // 16 (padded)
    float* t1   = ws + 2064;          // 64
    float* gi   = ws + 2128;          // B*3H   = 196608
    float* rnn  = gi   + (long)B_ * H3_;   // B*H = 65536
    float* hbuf = rnn  + (long)B_ * H_;    // 2*H = 2048
    float* sc   = hbuf + 2 * H_;           // B*S = 16384
    float* aw   = sc   + B_ * S_;          // B*S = 16384
    float* cin  = aw   + B_ * S_;          // B*2H = 131072
    float* cout = cin  + (long)B_ * H2_;   // B*H  = 65536
    unsigned* sync = (unsigned*)(cout + (long)B_ * H_);

    float* out_logits = (float*)d_out;                    // B*V
    float* out_hidden = out_logits + (long)B_ * V_;       // H
    float* out_attn   = out_hidden + H_;                  // B*S

    hipLaunchKernelGGL(reset_sync_kernel, dim3(1), dim3(32), 0, stream, sync);
    hipLaunchKernelGGL(attn_u_kernel, dim3(9), dim3(256), 0, stream,
                       W_attn, b_attn, v_attn, u, c0);
    // gi = emb_table[input_seq] @ W_ih^T + b_ih    (gather fused into GEMM)
    hipLaunchKernelGGL(gemm_wmma_f32, dim3(H3_ / 16, B_ / 16), dim3(32), 0, stream,
                       emb, H_, W_ih, H_, b_ih, gi, H3_, H_, 0, input_seq);
    hipLaunchKernelGGL(gru_scan_kernel, dim3(32), dim3(256), 0, stream,
                       gi, W_hh, b_hh, last_hidden, hbuf, rnn, out_hidden, sync);
    hipLaunchKernelGGL(t1_kernel, dim3(B_), dim3(32), 0, stream, u, rnn, t1);
    hipLaunchKernelGGL(scores_kernel, dim3((B_ * S_) / 8), dim3(256), 0, stream,
                       u + H_, enc, t1, c0, sc);
    hipLaunchKernelGGL(softmax_kernel, dim3(B_), dim3(256), 0, stream, sc, aw, out_attn);
    hipLaunchKernelGGL(context_kernel, dim3(B_ * 4), dim3(256), 0, stream, aw, enc, rnn, cin);
    // concat_output = tanh(concat_input @ W_concat^T + b_concat)
    hipLaunchKernelGGL(gemm_wmma_f32, dim3(H_ / 16, B_ / 16), dim3(32), 0, stream,
                       cin, H2_, W_concat, H2_, b_concat, cout, H_, H2_, 1, (const int*)nullptr);
    // output = concat_output @ W_out^T + b_out   (N = 10000 = 625*16, tile-exact)
    hipLaunchKernelGGL(gemm_wmma_f32, dim3(V_ / 16, B_ / 16), dim3(32), 0, stream,
                       cout, H_, W_out, H_, b_out, out_logits, V_, H_, 0, (const int*)nullptr);
}